// SS2D6_50345606644093
// MI455X (gfx1250) — compile-verified
//
#include <hip/hip_runtime.h>
#include <math.h>

// ---------------------------------------------------------------------------
// SS2D (VMamba selective-scan 2D) for MI455X / gfx1250, wave32 + WMMA f32.
// Shapes: B=1, H=W=64 (L=4096), d_model=64, di=128, N=16, K=6 dirs, dt_rank=4.
// ---------------------------------------------------------------------------

typedef __attribute__((ext_vector_type(2))) float v2f;
typedef __attribute__((ext_vector_type(8))) float v8f;

#define LQ      4096    // H*W
#define DI      128     // d_inner
#define NST     16      // state dim N
#define KDIR    6       // scan directions
#define NCHUNK  64      // chunks along L
#define CHLEN   64      // chunk length

__device__ __forceinline__ float softplus_f(float x) {
  return (x > 20.0f) ? x : log1pf(expf(x));
}
__device__ __forceinline__ float sigmoid_f(float x) {
  return 1.0f / (1.0f + expf(-x));
}

// One K=4 step of D = A(16x4,f32) * B(4x16,f32) + C(16x16,f32).
// A layout (ISA 7.12.2): lanes 0-15 M=lane {VGPR0:K=0, VGPR1:K=1},
// lanes 16-31 M=lane-16 {VGPR0:K=2, VGPR1:K=3}.  B mirrors with N=lane%16.
__device__ __forceinline__ v8f wmma_f32_step(v2f a, v2f b, v8f c) {
  return __builtin_amdgcn_wmma_f32_16x16x4_f32(
      /*neg_a=*/false, a, /*neg_b=*/false, b,
      /*c_mod=*/(short)0, c, /*reuse_a=*/false, /*reuse_b=*/false);
}

// ---------------------------------------------------------------------------
// Permutation tables: perm[k][l] = memory index read for scan position l;
// iw[k][p] = scan position whose output is accumulated at memory index p.
// k0: identity  k1: transpose  k2/k3: reversed k0/k1
// k4: anti-diagonal (stable argsort of i+j)  k5: flipW then anti-diagonal,
//     but written back through the FULL flip (matches the reference exactly).
// ---------------------------------------------------------------------------
__global__ void k_setup_perms(int* __restrict__ perm, int* __restrict__ iw) {
  int l = blockIdx.x * blockDim.x + threadIdx.x;
  if (l >= LQ) return;
  int t = (l & 63) * 64 + (l >> 6);                 // transpose map (involution)
  perm[0 * LQ + l] = l;
  perm[1 * LQ + l] = t;
  perm[2 * LQ + l] = (LQ - 1) - l;
  int lr = (LQ - 1) - l;
  perm[3 * LQ + l] = (lr & 63) * 64 + (lr >> 6);
  int i = l >> 6, j = l & 63;
  int s = i + j;                                    // anti-diagonal id, 0..126
  int off  = (s < 64) ? (s * (s + 1)) / 2
                      : (LQ - ((127 - s) * (128 - s)) / 2);
  int imin = (s > 63) ? (s - 63) : 0;
  int pos  = off + (i - imin);                      // scan position of (i,j)
  perm[4 * LQ + pos] = l;                           // scatter (unique)
  perm[5 * LQ + pos] = i * 64 + (63 - j);           // flip-W read order
  iw[0 * LQ + l] = l;
  iw[1 * LQ + l] = t;
  iw[2 * LQ + l] = (LQ - 1) - l;
  iw[3 * LQ + l] = (LQ - 1) - t;
  iw[4 * LQ + l] = pos;
  iw[5 * LQ + ((LQ - 1) - l)] = pos;                // full-flip write-back
}

// ---------------------------------------------------------------------------
// in_proj: xz[l][o] = sum_c x[l][c] * W_in[o][c]; split into xcin (d,l) / z (d,l)
// M=4096, N=256, K=64.  One wave per 16x16 tile.
// ---------------------------------------------------------------------------
__global__ void k_gemm_inproj(const float* __restrict__ x,
                              const float* __restrict__ Win,
                              float* __restrict__ xcin, float* __restrict__ zbuf) {
  int wave = (blockIdx.x * blockDim.x + threadIdx.x) >> 5;
  int lane = threadIdx.x & 31;
  int hi = lane >> 4, li = lane & 15;
  int tm = wave >> 4, tn = wave & 15;               // 256 x 16 tiles
  int m0 = tm * 16, n0 = tn * 16;
  int m = m0 + li, n = n0 + li;
  v8f acc = {};
  for (int k0 = 0; k0 < 64; k0 += 4) {
    int kk = k0 + 2 * hi;
    v2f a, b;
    a.x = x[m * 64 + kk];     a.y = x[m * 64 + kk + 1];
    b.x = Win[n * 64 + kk];   b.y = Win[n * 64 + kk + 1];   // B[c][o]=W_in[o][c]
    acc = wmma_f32_step(a, b, acc);
  }
#pragma unroll
  for (int r = 0; r < 8; ++r) {
    int row = m0 + r + 8 * hi;                      // C/D layout
    int o = n0 + li;
    float v = acc[r];
    if (o < DI) xcin[o * LQ + row] = v;
    else        zbuf[(o - DI) * LQ + row] = v;
  }
}

// ---------------------------------------------------------------------------
// Depthwise 3x3 conv (SAME, cross-correlation) + bias + SiLU.  (d,l) layout.
// ---------------------------------------------------------------------------
__global__ void k_conv_silu(const float* __restrict__ xcin,
                            const float* __restrict__ cw,
                            const float* __restrict__ cb,
                            float* __restrict__ xc) {
  int idx = blockIdx.x * 256 + threadIdx.x;         // 128*4096 threads
  int d = idx >> 12, l = idx & (LQ - 1);
  int h = l >> 6, w = l & 63;
  float acc = cb[d];
  const float* src = xcin + d * LQ;
  const float* wt  = cw + d * 9;
#pragma unroll
  for (int kh = 0; kh < 3; ++kh) {
    int hh = h + kh - 1;
    if (hh < 0 || hh > 63) continue;
#pragma unroll
    for (int kw = 0; kw < 3; ++kw) {
      int ww = w + kw - 1;
      if (ww < 0 || ww > 63) continue;
      acc += src[hh * 64 + ww] * wt[kh * 3 + kw];
    }
  }
  xc[d * LQ + l] = acc * sigmoid_f(acc);
}

// ---------------------------------------------------------------------------
// x_proj per direction: xdbl[k][c][l] = sum_d xpw[k][c][d] * xc[d][perm_k(l)]
// M=48 (36 padded), N=4096, K=128.
// ---------------------------------------------------------------------------
__global__ void k_gemm_xproj(const float* __restrict__ xpw,
                             const float* __restrict__ xc,
                             const int* __restrict__ perm,
                             float* __restrict__ xdbl) {
  int wave = (blockIdx.x * blockDim.x + threadIdx.x) >> 5;
  int lane = threadIdx.x & 31;
  int hi = lane >> 4, li = lane & 15;
  int k = wave / 768;
  int rem = wave - k * 768;
  int tm = rem >> 8, tn = rem & 255;                // 3 x 256 tiles
  int m0 = tm * 16, n0 = tn * 16;
  int mRow = m0 + li;
  int p = perm[k * LQ + (n0 + li)];                 // gather index, hoisted
  const float* Ak = xpw + k * 36 * DI;
  bool okm = (mRow < 36);
  v8f acc = {};
  for (int k0 = 0; k0 < DI; k0 += 4) {
    int kk = k0 + 2 * hi;
    v2f a, b;
    a.x = okm ? Ak[mRow * DI + kk]     : 0.0f;
    a.y = okm ? Ak[mRow * DI + kk + 1] : 0.0f;
    b.x = xc[kk * LQ + p];
    b.y = xc[(kk + 1) * LQ + p];
    acc = wmma_f32_step(a, b, acc);
  }
#pragma unroll
  for (int r = 0; r < 8; ++r) {
    int c = m0 + r + 8 * hi;
    if (c < 36) xdbl[(k * 36 + c) * LQ + (n0 + li)] = acc[r];
  }
}

// ---------------------------------------------------------------------------
// Chunked linear scan, phase 1: per (k,d,n,chunk) compute P = prod(dA) and
// S = local scan with x0=0.  16 lanes (one per n) share a (k,d,chunk).
// ---------------------------------------------------------------------------
__global__ void k_scan_phase1(const float* __restrict__ xdbl,
                              const float* __restrict__ xc,
                              const int* __restrict__ perm,
                              const float* __restrict__ dtw,
                              const float* __restrict__ dtb,
                              const float* __restrict__ A_logs,
                              float* __restrict__ chP, float* __restrict__ chS) {
  int g = (blockIdx.x * blockDim.x + threadIdx.x) >> 4;   // 49152 groups
  int n = threadIdx.x & 15;
  int c = g & 63, d = (g >> 6) & 127, k = g >> 13;
  int kd = k * DI + d;
  float An = -expf(A_logs[kd * NST + n]);
  float bias = dtb[kd];
  float w0 = dtw[kd * 4 + 0], w1 = dtw[kd * 4 + 1];
  float w2 = dtw[kd * 4 + 2], w3 = dtw[kd * 4 + 3];
  const float* xd = xdbl + k * 36 * LQ;
  const float* uB = xc + d * LQ;
  const int* pk = perm + k * LQ;
  float P = 1.0f, S = 0.0f;
  int l0 = c * CHLEN;
  for (int t = 0; t < CHLEN; ++t) {
    int l = l0 + t;
    float dts = w0 * xd[0 * LQ + l] + w1 * xd[1 * LQ + l] +
                w2 * xd[2 * LQ + l] + w3 * xd[3 * LQ + l] + bias;
    float delta = softplus_f(dts);
    float u = uB[pk[l]];
    float dA = expf(delta * An);
    float dBu = delta * xd[(4 + n) * LQ + l] * u;
    P *= dA;
    S = dA * S + dBu;
  }
  int idx = kd * NST + n;
  chP[idx * NCHUNK + c] = P;
  chS[idx * NCHUNK + c] = S;
}

// Phase 2: serial combine over the 64 chunks (12288 independent threads).
__global__ void k_scan_phase2(const float* __restrict__ chP,
                              const float* __restrict__ chS,
                              float* __restrict__ x0c) {
  int idx = blockIdx.x * 256 + threadIdx.x;         // 0..12287
  float x0 = 0.0f;
  for (int c = 0; c < NCHUNK; ++c) {
    x0c[idx * NCHUNK + c] = x0;
    x0 = chP[idx * NCHUNK + c] * x0 + chS[idx * NCHUNK + c];
  }
}

// Phase 3: replay chunk from correct initial state; y = sum_n x[n]*C[n] + u*D,
// reduced across the 16 state lanes with wave32 shfl_xor (width 16).
__global__ void k_scan_phase3(const float* __restrict__ xdbl,
                              const float* __restrict__ xc,
                              const int* __restrict__ perm,
                              const float* __restrict__ dtw,
                              const float* __restrict__ dtb,
                              const float* __restrict__ A_logs,
                              const float* __restrict__ Dsv,
                              const float* __restrict__ x0c,
                              float* __restrict__ outy) {
  int g = (blockIdx.x * blockDim.x + threadIdx.x) >> 4;
  int n = threadIdx.x & 15;
  int c = g & 63, d = (g >> 6) & 127, k = g >> 13;
  int kd = k * DI + d;
  float An = -expf(A_logs[kd * NST + n]);
  float bias = dtb[kd];
  float w0 = dtw[kd * 4 + 0], w1 = dtw[kd * 4 + 1];
  float w2 = dtw[kd * 4 + 2], w3 = dtw[kd * 4 + 3];
  const float* xd = xdbl + k * 36 * LQ;
  const float* uB = xc + d * LQ;
  const int* pk = perm + k * LQ;
  float xs = x0c[(kd * NST + n) * NCHUNK + c];
  float Dv = Dsv[kd];
  int l0 = c * CHLEN;
  for (int t = 0; t < CHLEN; ++t) {
    int l = l0 + t;
    float dts = w0 * xd[0 * LQ + l] + w1 * xd[1 * LQ + l] +
                w2 * xd[2 * LQ + l] + w3 * xd[3 * LQ + l] + bias;
    float delta = softplus_f(dts);
    float u = uB[pk[l]];
    float dA = expf(delta * An);
    float dBu = delta * xd[(4 + n) * LQ + l] * u;
    xs = dA * xs + dBu;
    float y = xs * xd[(20 + n) * LQ + l];           // C contraction term
    y += __shfl_xor(y, 8, 16);
    y += __shfl_xor(y, 4, 16);
    y += __shfl_xor(y, 2, 16);
    y += __shfl_xor(y, 1, 16);
    if (n == 0) outy[kd * LQ + l] = y + u * Dv;
  }
}

// ---------------------------------------------------------------------------
// Gather the 6 direction outputs at each spatial position p, LayerNorm over
// di=128, gate by silu(z), store yg[p][d] (row-major for the out_proj GEMM).
// ---------------------------------------------------------------------------
__global__ void k_final_ln(const float* __restrict__ outy,
                           const int* __restrict__ iw,
                           const float* __restrict__ zbuf,
                           const float* __restrict__ gamma,
                           const float* __restrict__ beta,
                           float* __restrict__ yg) {
  __shared__ float sm[DI];
  int p = blockIdx.x;
  int d = threadIdx.x;
  float y = 0.0f;
#pragma unroll
  for (int k = 0; k < KDIR; ++k)
    y += outy[(k * DI + d) * LQ + iw[k * LQ + p]];
  sm[d] = y; __syncthreads();
  for (int s = 64; s > 0; s >>= 1) { if (d < s) sm[d] += sm[d + s]; __syncthreads(); }
  float mu = sm[0] * (1.0f / DI);
  __syncthreads();
  float diff = y - mu;
  sm[d] = diff * diff; __syncthreads();
  for (int s = 64; s > 0; s >>= 1) { if (d < s) sm[d] += sm[d + s]; __syncthreads(); }
  float var = sm[0] * (1.0f / DI);
  float yn = diff * rsqrtf(var + 1e-5f) * gamma[d] + beta[d];
  float zv = zbuf[d * LQ + p];
  yg[p * DI + d] = yn * (zv * sigmoid_f(zv));
}

// ---------------------------------------------------------------------------
// out_proj: out[l][o] = sum_d yg[l][d] * W_out[o][d].  M=4096, N=64, K=128.
// ---------------------------------------------------------------------------
__global__ void k_gemm_out(const float* __restrict__ yg,
                           const float* __restrict__ Wout,
                           float* __restrict__ out) {
  int wave = (blockIdx.x * blockDim.x + threadIdx.x) >> 5;
  int lane = threadIdx.x & 31;
  int hi = lane >> 4, li = lane & 15;
  int tm = wave >> 2, tn = wave & 3;                // 256 x 4 tiles
  int m0 = tm * 16, n0 = tn * 16;
  int m = m0 + li, n = n0 + li;
  v8f acc = {};
  for (int k0 = 0; k0 < DI; k0 += 4) {
    int kk = k0 + 2 * hi;
    v2f a, b;
    a.x = yg[m * DI + kk];    a.y = yg[m * DI + kk + 1];
    b.x = Wout[n * DI + kk];  b.y = Wout[n * DI + kk + 1];
    acc = wmma_f32_step(a, b, acc);
  }
#pragma unroll
  for (int r = 0; r < 8; ++r)
    out[(m0 + r + 8 * hi) * 64 + (n0 + li)] = acc[r];
}

// ---------------------------------------------------------------------------
extern "C" void kernel_launch(void* const* d_in, const int* in_sizes, int n_in,
                              void* d_out, int out_size, void* d_ws, size_t ws_size,
                              hipStream_t stream) {
  (void)in_sizes; (void)n_in; (void)out_size; (void)ws_size;
  const float* x      = (const float*)d_in[0];   // (1,64,64,64)
  const float* W_in   = (const float*)d_in[1];   // (256,64)
  const float* conv_w = (const float*)d_in[2];   // (128,1,3,3)
  const float* conv_b = (const float*)d_in[3];   // (128)
  const float* xpw    = (const float*)d_in[4];   // (6,36,128)
  const float* dt_w   = (const float*)d_in[5];   // (6,128,4)
  const float* dt_b   = (const float*)d_in[6];   // (6,128)
  const float* A_logs = (const float*)d_in[7];   // (768,16)
  const float* Ds     = (const float*)d_in[8];   // (768)
  const float* ln_g   = (const float*)d_in[9];   // (128)
  const float* ln_b   = (const float*)d_in[10];  // (128)
  const float* W_out  = (const float*)d_in[11];  // (64,128)
  float* out = (float*)d_out;

  // Workspace layout (~34.1 MB total).
  int*   perm = (int*)d_ws;                        // 6*4096 ints
  int*   iw   = perm + KDIR * LQ;                  // 6*4096 ints
  float* xcin = (float*)(iw + KDIR * LQ);          // 128*4096
  float* zbuf = xcin + DI * LQ;                    // 128*4096
  float* xc   = zbuf + DI * LQ;                    // 128*4096
  float* xdbl = xc + DI * LQ;                      // 6*36*4096
  float* chP  = xdbl + KDIR * 36 * LQ;             // 6*128*16*64
  float* chS  = chP + KDIR * DI * NST * NCHUNK;
  float* x0c  = chS + KDIR * DI * NST * NCHUNK;
  float* outy = x0c + KDIR * DI * NST * NCHUNK;    // 6*128*4096
  float* yg   = outy + KDIR * DI * LQ;             // 4096*128

  k_setup_perms <<<16, 256, 0, stream>>>(perm, iw);
  k_gemm_inproj <<<512, 256, 0, stream>>>(x, W_in, xcin, zbuf);     // 4096 waves
  k_conv_silu   <<<2048, 256, 0, stream>>>(xcin, conv_w, conv_b, xc);
  k_gemm_xproj  <<<576, 256, 0, stream>>>(xpw, xc, perm, xdbl);     // 4608 waves
  k_scan_phase1 <<<3072, 256, 0, stream>>>(xdbl, xc, perm, dt_w, dt_b, A_logs, chP, chS);
  k_scan_phase2 <<<48, 256, 0, stream>>>(chP, chS, x0c);
  k_scan_phase3 <<<3072, 256, 0, stream>>>(xdbl, xc, perm, dt_w, dt_b, A_logs, Ds, x0c, outy);
  k_final_ln    <<<4096, 128, 0, stream>>>(outy, iw, zbuf, ln_g, ln_b, yg);
  k_gemm_out    <<<128, 256, 0, stream>>>(yg, W_out, out);          // 1024 waves
}